// saivrNet_42752104465077
// MI455X (gfx1250) — compile-verified
//
#include <hip/hip_runtime.h>

typedef __attribute__((ext_vector_type(2))) float v2f;
typedef __attribute__((ext_vector_type(8))) float v8f;

#define N_LAYERS 10
#define TILES_PER_WAVE 4

__device__ __forceinline__ float sigmoidf_fast(float v) {
#if __has_builtin(__builtin_amdgcn_tanhf)
    // sigmoid(x) = 0.5*tanh(x/2) + 0.5 : 1 mul + 1 v_tanh_f32 + 1 fma
    return fmaf(0.5f, __builtin_amdgcn_tanhf(v * 0.5f), 0.5f);
#else
    return __builtin_amdgcn_rcpf(1.0f + __expf(-v));
#endif
}

// Weight tiles use a DUPLICATED-ROW layout: tile row 8 is a copy of row 2
// (and bias row 8 = b[2]).  Then the WMMA D tile directly matches the next
// WMMA's B layout:
//   D vgpr0: lanes 0-15 = feature0 (K0), lanes 16-31 = feature2 (K2)
//   D vgpr1: lanes 0-15 = feature1 (K1), lanes 16-31 = junk (K3, killed by
//            the zero K3 column of the A tile)
// so inter-layer repacking is just B = {acc[0], acc[1]} — no shuffles.

__global__ __launch_bounds__(256)
void saivr_wmma_kernel(const float* __restrict__ x,
                       const float* __restrict__ W_in,  const float* __restrict__ b_in,
                       const float* __restrict__ W_h,   const float* __restrict__ b_h,
                       const float* __restrict__ W_out, const float* __restrict__ b_out,
                       float* __restrict__ out, int N, int nTiles)
{
    const int lane          = threadIdx.x & 31;
    const int waveInBlock   = threadIdx.x >> 5;
    const int wavesPerBlock = blockDim.x >> 5;
    const long wave         = (long)blockIdx.x * wavesPerBlock + waveInBlock;

    const int  m  = lane & 15;          // tile row (A) / tile col (B,D) in half-wave
    const bool lo = lane < 16;          // lanes 0-15: K0/K1 ; lanes 16-31: K2/K3
    const int  kb = lo ? 0 : 2;

    // duplicated-row mapping: tile rows {0,1,2} -> W rows {0,1,2}; tile row 8 -> W row 2
    const bool wvalid = (m < 3) || (m == 8);
    const int  rr     = (m < 3) ? m : 2;

    // ---- A tiles (16x4 f32 A-layout) ----
    v2f a_in_lo, a_in_hi, a_h, a_o;
    // W_in: 3x8 row-major, K=8 split into [0..3] | [4..7]
    a_in_lo.x = wvalid ? W_in[rr * 8 + kb    ] : 0.0f;
    a_in_lo.y = (wvalid && lo) ? W_in[rr * 8 + 1] : 0.0f;      // K1 (lo) / K3=0 (hi)... K split below
    a_in_hi.x = wvalid ? W_in[rr * 8 + kb + 4] : 0.0f;
    a_in_hi.y = (wvalid && lo) ? W_in[rr * 8 + 5] : 0.0f;
    // fix: for the lo-K chunk the hi half-wave carries K2/K3 = cols 2,3 of that chunk
    if (!lo) {
        a_in_lo.y = wvalid ? W_in[rr * 8 + 3] : 0.0f;          // K3 of chunk0 = col 3
        a_in_hi.y = wvalid ? W_in[rr * 8 + 7] : 0.0f;          // K3 of chunk1 = col 7
    }
    // W_h: 3x3 row-major, K padded 3->4 (K3 col = 0)
    a_h.x = wvalid ? W_h[rr * 3 + kb] : 0.0f;
    a_h.y = (wvalid && lo) ? W_h[rr * 3 + 1] : 0.0f;           // hi half K3 = 0
    // W_out: 5x3 row-major (no duplication needed; outputs read from lanes 0-15)
    a_o.x = (m < 5) ? W_out[m * 3 + kb] : 0.0f;
    a_o.y = (m < 5 && lo) ? W_out[m * 3 + 1] : 0.0f;

    // ---- C tiles (bias): vgpr r = row M=r (lanes 0-15) / row M=r+8 (lanes 16-31) ----
    v8f c_in = {}, c_h = {}, c_o = {};
    c_in[0] = lo ? b_in[0] : b_in[2];      // row 8 = bias of duplicated feature 2
    c_in[1] = lo ? b_in[1] : 0.0f;
    c_h [0] = lo ? b_h [0] : b_h [2];
    c_h [1] = lo ? b_h [1] : 0.0f;
    if (lo) {
        c_o[0] = b_out[0]; c_o[1] = b_out[1]; c_o[2] = b_out[2];
        c_o[3] = b_out[3]; c_o[4] = b_out[4];
    }

    for (int i = 0; i < TILES_PER_WAVE; ++i) {
        long t = wave * TILES_PER_WAVE + i;
        if (t >= nTiles) break;                    // uniform per wave
        long row0 = t * 16;
        long rowm = row0 + m;
        long rclamp = (rowm < N) ? rowm : (long)(N - 1);

        const float* xr = x + rclamp * 8 + kb;
        __builtin_prefetch(xr + 16 * 8, 0, 0);     // speculative prefetch of next tile

        v2f b_l = *(const v2f*)(xr);               // chunk0: K = kb, kb+1
        v2f b_u = *(const v2f*)(xr + 4);           // chunk1: K = kb+4, kb+5

        // ---- input layer (raw pre-activation, K=8 via two chained WMMAs) ----
        v8f acc;
        acc = __builtin_amdgcn_wmma_f32_16x16x4_f32(false, a_in_lo, false, b_l,
                                                    (short)0, c_in, false, false);
        acc = __builtin_amdgcn_wmma_f32_16x16x4_f32(false, a_in_hi, false, b_u,
                                                    (short)0, acc, false, false);

        // ---- 10 hidden layers: h = sigmoid(W_h @ h + b_h) ----
#pragma unroll
        for (int l = 0; l < N_LAYERS; ++l) {
            v2f bh; bh.x = acc[0]; bh.y = acc[1];          // direct repack (see layout note)
            acc = __builtin_amdgcn_wmma_f32_16x16x4_f32(false, a_h, false, bh,
                                                        (short)0, c_h, false, false);
            acc[0] = sigmoidf_fast(acc[0]);                // feat0 (lo) + feat2 (hi)
            acc[1] = sigmoidf_fast(acc[1]);                // feat1 (lo) + junk (hi, x0 later)
        }

        // ---- output layer: out = W_out @ h + b_out ----
        v2f bh; bh.x = acc[0]; bh.y = acc[1];
        v8f o = __builtin_amdgcn_wmma_f32_16x16x4_f32(false, a_o, false, bh,
                                                      (short)0, c_o, false, false);
        if (lo && rowm < N) {
            // d_out = [s(N), a(N), i(N), v(N), r(N)] concatenated
            out[0L * N + rowm] = o[0];
            out[1L * N + rowm] = o[1];
            out[2L * N + rowm] = o[2];
            out[3L * N + rowm] = o[3];
            out[4L * N + rowm] = o[4];
        }
    }
}

extern "C" void kernel_launch(void* const* d_in, const int* in_sizes, int n_in,
                              void* d_out, int out_size, void* d_ws, size_t ws_size,
                              hipStream_t stream) {
    const float* x     = (const float*)d_in[0];
    const float* W_in  = (const float*)d_in[1];
    const float* b_in  = (const float*)d_in[2];
    const float* W_h   = (const float*)d_in[3];
    const float* b_h   = (const float*)d_in[4];
    const float* W_out = (const float*)d_in[5];
    const float* b_out = (const float*)d_in[6];
    float* out = (float*)d_out;

    int N      = in_sizes[0] / 8;
    int nTiles = (N + 15) / 16;

    const int block         = 256;                  // 8 waves (wave32)
    const int wavesPerBlock = block / 32;
    const int tilesPerBlock = wavesPerBlock * TILES_PER_WAVE;
    int grid = (nTiles + tilesPerBlock - 1) / tilesPerBlock;

    saivr_wmma_kernel<<<grid, block, 0, stream>>>(x, W_in, b_in, W_h, b_h,
                                                  W_out, b_out, out, N, nTiles);
}